// TopKPooling_59107339927786
// MI455X (gfx1250) — compile-verified
//
#include <hip/hip_runtime.h>
#include <math.h>

#define NUM_GRAPHS 100
#define NPG        1000
#define NNODES     100000
#define NEDGES     3200000
#define DFEAT      256
#define DEDGE      16
#define KSEL       500
#define NSEL       50000

// output offsets (float elements), concatenated in reference return order
#define OUT_X      0ULL          // 50000*256      = 12,800,000
#define OUT_EI     12800000ULL   // 2*3,200,000    =  6,400,000
#define OUT_EA     19200000ULL   // 3,200,000*16   = 51,200,000
#define OUT_BATCH  70400000ULL   // 50,000
#define OUT_PERM   70450000ULL   // 50,000
#define OUT_SCORE  70500000ULL   // 50,000

// workspace byte offsets
#define WS_INV     0
#define WS_SCORES  256        // float[100000]
#define WS_PERM    400384     // int[50000]
#define WS_SELSC   600448     // float[50000]
#define WS_NEWIDS  800512     // int[100000]
#define WS_MASK    1200640    // u8[3200000]

typedef float v2f __attribute__((ext_vector_type(2)));
typedef float v8f __attribute__((ext_vector_type(8)));

// ---------------- 1 / ||w|| ----------------
__global__ __launch_bounds__(256) void norm_kernel(const float* __restrict__ w,
                                                   float* __restrict__ invn) {
    __shared__ float red[256];
    const int t = threadIdx.x;
    const float v = w[t];
    red[t] = v * v;
    __syncthreads();
    for (int s = 128; s > 0; s >>= 1) {
        if (t < s) red[t] += red[t + s];
        __syncthreads();
    }
    if (t == 0) invn[0] = 1.0f / sqrtf(red[0]);
}

// ---------------- score = tanh((x@w)/||w||) via V_WMMA_F32_16X16X4_F32 ----------------
// One wave handles 16 node rows. A = 16x4 slice of x (lane = row M, lane>>4 selects K half).
// B = w[k0..k0+3] broadcast across all 16 columns -> every D column equals the GEMV result.
__global__ __launch_bounds__(256) void score_kernel(const float* __restrict__ x,
                                                    const float* __restrict__ w,
                                                    const float* __restrict__ invn,
                                                    float* __restrict__ scores) {
    const int tid  = threadIdx.x;
    const int lane = tid & 31;
    const int wv   = tid >> 5;
    const int base = (blockIdx.x * 8 + wv) * 16;
    if (base >= NNODES) return;                 // uniform per wave: EXEC stays all-ones

    const int row  = lane & 15;
    const int koff = (lane >> 4) << 1;          // lanes 16-31 hold K+2,K+3
    const float* __restrict__ xrow = x + (size_t)(base + row) * DFEAT + koff;
    const float* __restrict__ wp   = w + koff;

    v8f acc = {};
#pragma unroll 4
    for (int k0 = 0; k0 < DFEAT; k0 += 4) {
        v2f a = *(const v2f*)(xrow + k0);        // {x[row,k0+koff], x[row,k0+koff+1]}
        v2f b = *(const v2f*)(wp + k0);          // {w[k0+koff],     w[k0+koff+1]}
        acc = __builtin_amdgcn_wmma_f32_16x16x4_f32(
            false, a, false, b, (short)0, acc, false, false);
    }

    const float inv = invn[0];
    // D column 0: lane 0 has rows 0..7 in acc[0..7], lane 16 has rows 8..15
    if (lane == 0) {
#pragma unroll
        for (int m = 0; m < 8; ++m) scores[base + m] = tanhf(acc[m] * inv);
    } else if (lane == 16) {
#pragma unroll
        for (int m = 0; m < 8; ++m) scores[base + 8 + m] = tanhf(acc[m] * inv);
    }
}

// ---------------- per-graph top-k: bitonic sort 1024 in LDS ----------------
// Order: descending score, ties broken by ascending index (stable like lax.top_k).
__global__ __launch_bounds__(1024) void topk_kernel(const float* __restrict__ scores,
                                                    int* __restrict__ perm,
                                                    float* __restrict__ selsc,
                                                    int* __restrict__ new_ids,
                                                    float* __restrict__ out) {
    __shared__ float sv[1024];
    __shared__ int   si[1024];
    const int tid = threadIdx.x;
    const int g   = blockIdx.x;

    sv[tid] = (tid < NPG) ? scores[g * NPG + tid] : -INFINITY;
    si[tid] = tid;
    __syncthreads();

    for (int k = 2; k <= 1024; k <<= 1) {
        for (int j = k >> 1; j > 0; j >>= 1) {
            const int ixj = tid ^ j;
            if (ixj > tid) {
                const float v0 = sv[tid], v1 = sv[ixj];
                const int   i0 = si[tid], i1 = si[ixj];
                // "tid element precedes ixj element" in our total order
                const bool p01 = (v0 > v1) || (v0 == v1 && i0 < i1);
                const bool doswap = ((tid & k) == 0) ? !p01 : p01;
                if (doswap) { sv[tid] = v1; sv[ixj] = v0; si[tid] = i1; si[ixj] = i0; }
            }
            __syncthreads();
        }
    }

    if (tid < NPG) {                      // si[0..999] is a permutation of 0..999
        const int node = si[tid];
        new_ids[g * NPG + node] = (tid < KSEL) ? (g * KSEL + tid) : -1;
    }
    if (tid < KSEL) {
        const int o     = g * KSEL + tid;
        const int gnode = g * NPG + si[tid];
        perm[o]  = gnode;
        selsc[o] = sv[tid];
        out[OUT_PERM  + o] = (float)gnode;
        out[OUT_SCORE + o] = sv[tid];
        out[OUT_BATCH + o] = (float)g;
    }
}

// ---------------- x_new = x[perm] * sel_score ----------------
__global__ __launch_bounds__(256) void gather_kernel(const float* __restrict__ x,
                                                     const int* __restrict__ perm,
                                                     const float* __restrict__ selsc,
                                                     float* __restrict__ out_x) {
    const int row = blockIdx.x;
    const int t   = threadIdx.x;
    const int p   = perm[row];
    const float s = selsc[row];
    out_x[(size_t)row * DFEAT + t] = x[(size_t)p * DFEAT + t] * s;
}

// ---------------- edge remap + mask ----------------
__global__ __launch_bounds__(256) void edge_kernel(const int* __restrict__ ei,
                                                   const int* __restrict__ new_ids,
                                                   float* __restrict__ out_ei,
                                                   unsigned char* __restrict__ mask) {
    const int e = blockIdx.x * 256 + threadIdx.x;
    if (e >= NEDGES) return;
    const int s  = ei[e];
    const int d  = ei[NEDGES + e];
    const int ns = new_ids[s];
    const int nd = new_ids[d];
    const bool m = (ns >= 0) && (nd >= 0);
    out_ei[e]                   = (float)(m ? ns : -1);
    out_ei[(size_t)NEDGES + e]  = (float)(m ? nd : -1);
    mask[e] = m ? 1u : 0u;
}

// ---------------- edge_attr * mask (float4 stream, dominant bandwidth) ----------------
__global__ __launch_bounds__(256) void attr_kernel(const float4* __restrict__ ea,
                                                   const unsigned char* __restrict__ mask,
                                                   float4* __restrict__ out_ea) {
    const size_t gid = (size_t)blockIdx.x * 256 + threadIdx.x;   // NEDGES*4 float4s exactly
    const int e = (int)(gid >> 2);
    const float4 v = ea[gid];
    float4 o;
    if (mask[e]) { o = v; } else { o = make_float4(0.f, 0.f, 0.f, 0.f); }
    out_ea[gid] = o;
}

extern "C" void kernel_launch(void* const* d_in, const int* in_sizes, int n_in,
                              void* d_out, int out_size, void* d_ws, size_t ws_size,
                              hipStream_t stream) {
    (void)in_sizes; (void)n_in; (void)out_size; (void)ws_size;
    const float* x  = (const float*)d_in[0];
    const float* w  = (const float*)d_in[1];
    const float* ea = (const float*)d_in[2];
    const int*   ei = (const int*)d_in[3];
    // d_in[4] = batch: implied by uniform graph layout, not needed

    float* out = (float*)d_out;
    char*  ws  = (char*)d_ws;
    float*         invn    = (float*)(ws + WS_INV);
    float*         scores  = (float*)(ws + WS_SCORES);
    int*           perm    = (int*)(ws + WS_PERM);
    float*         selsc   = (float*)(ws + WS_SELSC);
    int*           newids  = (int*)(ws + WS_NEWIDS);
    unsigned char* mask    = (unsigned char*)(ws + WS_MASK);

    norm_kernel<<<1, 256, 0, stream>>>(w, invn);
    // 100000/16 = 6250 waves, 8 waves per block -> 782 blocks
    score_kernel<<<782, 256, 0, stream>>>(x, w, invn, scores);
    topk_kernel<<<NUM_GRAPHS, 1024, 0, stream>>>(scores, perm, selsc, newids, out);
    gather_kernel<<<NSEL, 256, 0, stream>>>(x, perm, selsc, out + OUT_X);
    edge_kernel<<<(NEDGES + 255) / 256, 256, 0, stream>>>(ei, newids, out + OUT_EI, mask);
    // NEDGES*16 floats / 4 per thread / 256 per block = 50000 blocks exactly
    attr_kernel<<<50000, 256, 0, stream>>>((const float4*)ea, mask, (float4*)(out + OUT_EA));
}